// GraphSAGENetwork_14431090115092
// MI455X (gfx1250) — compile-verified
//
#include <hip/hip_runtime.h>

#define N_NODES 50000
#define D_IN    128
#define D_HID   256
#define E_EDGES 800000

typedef __attribute__((ext_vector_type(2))) float v2f;
typedef __attribute__((ext_vector_type(8))) float v8f;

// Native CDNA5 L2-side float max atomic (no-return form, tracked by STOREcnt;
// drained by S_ENDPGM's implicit wait-idle). Guarantees we never fall back to
// a CAS loop: fmax is rounding-free, so this is bit-exact vs segment_max.
__device__ __forceinline__ void atomic_max_f32(float* addr, float val) {
  asm volatile("global_atomic_max_num_f32 %0, %1, off"
               :: "v"(addr), "v"(val) : "memory");
}

// ---------------------------------------------------------------------------
// Fill a float buffer with -inf (segment_max identity). n4 = count of float4.
// ---------------------------------------------------------------------------
__global__ void sage_fill_neginf(float* __restrict__ p, int n4) {
  int i = blockIdx.x * blockDim.x + threadIdx.x;
  if (i < n4) {
    const float ninf = -__builtin_inff();
    float4 v = make_float4(ninf, ninf, ninf, ninf);
    ((float4*)p)[i] = v;
  }
}

// ---------------------------------------------------------------------------
// Per-edge scatter-max: agg[dst] = max(agg[dst], feat[src]) elementwise.
// One thread handles one float4 chunk of one edge's message; a full wave
// shares one edge (wave-uniform index loads). All buffers are L2-resident.
// ---------------------------------------------------------------------------
template<int F>
__global__ void sage_scatter_max(const float* __restrict__ feat,
                                 const long long* __restrict__ ei,   // [2, E] int64
                                 float* __restrict__ agg) {
  constexpr int CH = F / 4;
  long long tid = (long long)blockIdx.x * blockDim.x + threadIdx.x;
  if (tid >= (long long)E_EDGES * CH) return;
  int e = (int)(tid / CH);
  int c = (int)(tid % CH);
  int src = (int)ei[e];
  int dst = (int)ei[E_EDGES + e];
  float4 v = ((const float4*)(feat + (size_t)src * F))[c];
  float* a = agg + (size_t)dst * F + (size_t)c * 4;
  atomic_max_f32(a + 0, v.x);
  atomic_max_f32(a + 1, v.y);
  atomic_max_f32(a + 2, v.z);
  atomic_max_f32(a + 3, v.w);
}

// ---------------------------------------------------------------------------
// Layer 1 GEMM: h = relu( agg1' @ Wl1.T + bl1 + x @ Wr1.T )
//   agg1' = agg1 with -inf (empty segment) replaced by 0, fused into A-load.
// Each wave computes a 16(M) x 64(N) output strip via v_wmma_f32_16x16x4_f32,
// reusing one A fragment across 4 N-tiles per K-step.
// ---------------------------------------------------------------------------
__global__ void sage_gemm1(const float* __restrict__ agg,  // [N,128], -inf = empty
                           const float* __restrict__ x,    // [N,128]
                           const float* __restrict__ Wl,   // [256,128]
                           const float* __restrict__ Wr,   // [256,128]
                           const float* __restrict__ bl,   // [256]
                           float* __restrict__ h)          // [N,256]
{
  constexpr int M_TILES = N_NODES / 16;   // 3125
  constexpr int NG      = D_HID / 64;     // 4 strips of 64 cols
  int wave = (blockIdx.x * blockDim.x + threadIdx.x) >> 5;
  if (wave >= M_TILES * NG) return;
  int lane = threadIdx.x & 31;
  int mt = wave / NG, ng = wave % NG;
  int m0 = mt * 16, n0 = ng * 64;
  int mrow = lane & 15;        // A row / B,C,D column-within-tile
  int hi   = lane >> 4;        // half-wave select
  int koff = hi * 2;           // K sub-offset for A/B fragments
  const float ninf = -__builtin_inff();

  v8f acc[4] = {};
  const float* aRow = agg + (size_t)(m0 + mrow) * D_IN;
  const float* xRow = x   + (size_t)(m0 + mrow) * D_IN;

  // term 1: agg1' @ Wl.T
  for (int k = 0; k < D_IN; k += 4) {
    v2f a = *(const v2f*)(aRow + k + koff);
    a.x = (a.x == ninf) ? 0.f : a.x;   // empty segment -> 0
    a.y = (a.y == ninf) ? 0.f : a.y;
#pragma unroll
    for (int nt = 0; nt < 4; ++nt) {
      const float* wRow = Wl + (size_t)(n0 + nt * 16 + mrow) * D_IN;
      v2f b = *(const v2f*)(wRow + k + koff);
      acc[nt] = __builtin_amdgcn_wmma_f32_16x16x4_f32(
          false, a, false, b, (short)0, acc[nt], false, false);
    }
  }
  // term 2: x @ Wr.T
  for (int k = 0; k < D_IN; k += 4) {
    v2f a = *(const v2f*)(xRow + k + koff);
#pragma unroll
    for (int nt = 0; nt < 4; ++nt) {
      const float* wRow = Wr + (size_t)(n0 + nt * 16 + mrow) * D_IN;
      v2f b = *(const v2f*)(wRow + k + koff);
      acc[nt] = __builtin_amdgcn_wmma_f32_16x16x4_f32(
          false, a, false, b, (short)0, acc[nt], false, false);
    }
  }
  // epilogue: bias + relu
#pragma unroll
  for (int nt = 0; nt < 4; ++nt) {
    int col = n0 + nt * 16 + mrow;
    float bias = bl[col];
#pragma unroll
    for (int r = 0; r < 8; ++r) {
      int row = m0 + hi * 8 + r;
      float v = acc[nt][r] + bias;
      h[(size_t)row * D_HID + col] = v > 0.f ? v : 0.f;
    }
  }
}

// ---------------------------------------------------------------------------
// Layer 2 GEMM: h2 = agg2' @ Wl2.T + bl2 + h @ Wr2.T + x   (residual fused)
// ---------------------------------------------------------------------------
__global__ void sage_gemm2(const float* __restrict__ agg,  // [N,256], -inf = empty
                           const float* __restrict__ h,    // [N,256]
                           const float* __restrict__ Wl,   // [128,256]
                           const float* __restrict__ Wr,   // [128,256]
                           const float* __restrict__ bl,   // [128]
                           const float* __restrict__ x,    // [N,128] residual
                           float* __restrict__ h2)         // [N,128]
{
  constexpr int M_TILES = N_NODES / 16;   // 3125
  constexpr int NG      = D_IN / 64;      // 2 strips of 64 cols
  int wave = (blockIdx.x * blockDim.x + threadIdx.x) >> 5;
  if (wave >= M_TILES * NG) return;
  int lane = threadIdx.x & 31;
  int mt = wave / NG, ng = wave % NG;
  int m0 = mt * 16, n0 = ng * 64;
  int mrow = lane & 15;
  int hi   = lane >> 4;
  int koff = hi * 2;
  const float ninf = -__builtin_inff();

  v8f acc[4] = {};
  const float* aRow = agg + (size_t)(m0 + mrow) * D_HID;
  const float* hRow = h   + (size_t)(m0 + mrow) * D_HID;

  for (int k = 0; k < D_HID; k += 4) {
    v2f a = *(const v2f*)(aRow + k + koff);
    a.x = (a.x == ninf) ? 0.f : a.x;
    a.y = (a.y == ninf) ? 0.f : a.y;
#pragma unroll
    for (int nt = 0; nt < 4; ++nt) {
      const float* wRow = Wl + (size_t)(n0 + nt * 16 + mrow) * D_HID;
      v2f b = *(const v2f*)(wRow + k + koff);
      acc[nt] = __builtin_amdgcn_wmma_f32_16x16x4_f32(
          false, a, false, b, (short)0, acc[nt], false, false);
    }
  }
  for (int k = 0; k < D_HID; k += 4) {
    v2f a = *(const v2f*)(hRow + k + koff);
#pragma unroll
    for (int nt = 0; nt < 4; ++nt) {
      const float* wRow = Wr + (size_t)(n0 + nt * 16 + mrow) * D_HID;
      v2f b = *(const v2f*)(wRow + k + koff);
      acc[nt] = __builtin_amdgcn_wmma_f32_16x16x4_f32(
          false, a, false, b, (short)0, acc[nt], false, false);
    }
  }
#pragma unroll
  for (int nt = 0; nt < 4; ++nt) {
    int col = n0 + nt * 16 + mrow;
    float bias = bl[col];
#pragma unroll
    for (int r = 0; r < 8; ++r) {
      int row = m0 + hi * 8 + r;
      h2[(size_t)row * D_IN + col] =
          acc[nt][r] + bias + x[(size_t)row * D_IN + col];
    }
  }
}

// ---------------------------------------------------------------------------
// LayerNorm over 128 features, one wave32 per node (float4 per lane).
// ---------------------------------------------------------------------------
__global__ void sage_layernorm(const float* __restrict__ h2,
                               const float* __restrict__ gamma,
                               const float* __restrict__ beta,
                               float* __restrict__ out) {
  int wave = (blockIdx.x * blockDim.x + threadIdx.x) >> 5;
  int lane = threadIdx.x & 31;
  if (wave >= N_NODES) return;
  float4 v = ((const float4*)(h2 + (size_t)wave * D_IN))[lane];
  float s = v.x + v.y + v.z + v.w;
#pragma unroll
  for (int off = 16; off > 0; off >>= 1) s += __shfl_xor(s, off, 32);
  float mu = s * (1.f / 128.f);
  float dx = v.x - mu, dy = v.y - mu, dz = v.z - mu, dw = v.w - mu;
  float q = dx * dx + dy * dy + dz * dz + dw * dw;
#pragma unroll
  for (int off = 16; off > 0; off >>= 1) q += __shfl_xor(q, off, 32);
  float rstd = rsqrtf(q * (1.f / 128.f) + 1e-5f);
  float4 g = ((const float4*)gamma)[lane];
  float4 b = ((const float4*)beta)[lane];
  float4 o;
  o.x = dx * rstd * g.x + b.x;
  o.y = dy * rstd * g.y + b.y;
  o.z = dz * rstd * g.z + b.z;
  o.w = dw * rstd * g.w + b.w;
  ((float4*)(out + (size_t)wave * D_IN))[lane] = o;
}

// ---------------------------------------------------------------------------
extern "C" void kernel_launch(void* const* d_in, const int* in_sizes, int n_in,
                              void* d_out, int out_size, void* d_ws, size_t ws_size,
                              hipStream_t stream) {
  const float*     x   = (const float*)d_in[0];
  const long long* ei  = (const long long*)d_in[1];   // int64 [2, E]
  const float*     Wl1 = (const float*)d_in[2];
  const float*     bl1 = (const float*)d_in[3];
  const float*     Wr1 = (const float*)d_in[4];
  const float*     Wl2 = (const float*)d_in[5];
  const float*     bl2 = (const float*)d_in[6];
  const float*     Wr2 = (const float*)d_in[7];
  const float*     gam = (const float*)d_in[8];
  const float*     bet = (const float*)d_in[9];
  float*           out = (float*)d_out;

  // Workspace layout (floats):
  float* ws   = (float*)d_ws;
  float* agg1 = ws;                                        // 50000*128 =  6.4M
  float* h    = ws + (size_t)N_NODES * D_IN;               // 50000*256 = 12.8M
  float* agg2 = h  + (size_t)N_NODES * D_HID;              // 50000*256 = 12.8M
  float* h2   = agg2 + (size_t)N_NODES * D_HID;            // 50000*128 =  6.4M

  const int TB = 256;

  // ---- layer 1 ----
  {
    int n4 = (N_NODES * D_IN) / 4;                          // 1.6M float4
    sage_fill_neginf<<<(n4 + TB - 1) / TB, TB, 0, stream>>>(agg1, n4);
  }
  {
    long long tot = (long long)E_EDGES * (D_IN / 4);        // 25.6M threads
    int blocks = (int)((tot + TB - 1) / TB);
    sage_scatter_max<D_IN><<<blocks, TB, 0, stream>>>(x, ei, agg1);
  }
  {
    int waves = (N_NODES / 16) * (D_HID / 64);              // 12500 waves
    int blocks = (waves * 32 + TB - 1) / TB;
    sage_gemm1<<<blocks, TB, 0, stream>>>(agg1, x, Wl1, Wr1, bl1, h);
  }

  // ---- layer 2 ----
  {
    int n4 = (N_NODES * D_HID) / 4;                         // 3.2M float4
    sage_fill_neginf<<<(n4 + TB - 1) / TB, TB, 0, stream>>>(agg2, n4);
  }
  {
    long long tot = (long long)E_EDGES * (D_HID / 4);       // 51.2M threads
    int blocks = (int)((tot + TB - 1) / TB);
    sage_scatter_max<D_HID><<<blocks, TB, 0, stream>>>(h, ei, agg2);
  }
  {
    int waves = (N_NODES / 16) * (D_IN / 64);               // 6250 waves
    int blocks = (waves * 32 + TB - 1) / TB;
    sage_gemm2<<<blocks, TB, 0, stream>>>(agg2, h, Wl2, Wr2, bl2, x, h2);
  }

  // ---- residual already fused; layernorm -> d_out ----
  {
    int blocks = (N_NODES * 32 + TB - 1) / TB;              // 1 wave / node
    sage_layernorm<<<blocks, TB, 0, stream>>>(h2, gam, bet, out);
  }
}